// QuadraFormer_65524021068234
// MI455X (gfx1250) — compile-verified
//
#include <hip/hip_runtime.h>
#include <math.h>

#define B_ 16
#define L_ 96
#define N_ 862
#define D_ 16
#define F_ 64
#define E_ 4
#define P_ 96
#define TS 17                 // padded tile row stride (bank-conflict free)
#define BN (B_ * N_)          // 13792
#define LD (L_ * D_)          // 1536
#define PI_F 3.14159265358979f

typedef float v2f __attribute__((ext_vector_type(2)));
typedef float v8f __attribute__((ext_vector_type(8)));

// ---- WMMA helpers: fp32 16x16x16 GEMM tile as 4x V_WMMA_F32_16X16X4_F32 ----
__device__ inline v8f wmma_k4(v2f a, v2f b, v8f c) {
  return __builtin_amdgcn_wmma_f32_16x16x4_f32(false, a, false, b, (short)0, c,
                                               false, false);
}

// C(16x16) += A(16x16) * B(16x16). A row-major stride `as`; B row-major stride
// `bs` (btrans -> use B^T, i.e. B[k][j] = Bm[j][k]).
__device__ inline v8f mm16(const float* A, int as, const float* Bm, int bs,
                           bool btrans, v8f acc, int lane) {
  const int M = lane & 15;
  const int hi = lane >> 4;
#pragma unroll
  for (int kk = 0; kk < 4; ++kk) {
    const int k0 = kk * 4 + 2 * hi;
    v2f a;
    a.x = A[M * as + k0];
    a.y = A[M * as + k0 + 1];
    v2f b;
    if (btrans) {
      b.x = Bm[M * bs + k0];
      b.y = Bm[M * bs + k0 + 1];
    } else {
      b.x = Bm[k0 * bs + M];
      b.y = Bm[(k0 + 1) * bs + M];
    }
    acc = wmma_k4(a, b, acc);
  }
  return acc;
}

// store C tile (VGPR r holds rows r and r+8) to LDS/global tile, stride ts_
__device__ inline void st16(float* T, int ts_, v8f acc, int lane, float mul) {
  const int cN = lane & 15;
  const int hi = lane >> 4;
#pragma unroll
  for (int r = 0; r < 8; ++r) T[(r + 8 * hi) * ts_ + cN] = acc[r] * mul;
}

__device__ inline float gelu_f(float x) {
  return 0.5f * x *
         (1.f + tanhf(0.7978845608028654f * (x + 0.044715f * x * x * x)));
}

// ---------------- K1: RevIN stats over time axis ----------------
__global__ void k_stats(const float* __restrict__ x, float* __restrict__ muv,
                        float* __restrict__ stdv) {
  int i = blockIdx.x * blockDim.x + threadIdx.x;
  if (i >= BN) return;
  int b = i / N_, n = i % N_;
  const float* xp = x + (size_t)b * L_ * N_ + n;
  float s = 0.f, s2 = 0.f;
  for (int l = 0; l < L_; ++l) {
    float v = xp[(size_t)l * N_];
    s += v;
    s2 += v * v;
  }
  float mu = s / L_;
  float var = s2 / L_ - mu * mu;
  muv[i] = mu;
  stdv[i] = sqrtf(var + 1e-5f);
}

// ---------------- K2: embedding (layout [B,N,L,D]) + gate channel ----------
__global__ void k_embed(const float* __restrict__ x, const float* __restrict__ muv,
                        const float* __restrict__ stdv, const float* __restrict__ sw,
                        const float* __restrict__ sb, float* __restrict__ emb,
                        float* __restrict__ g0) {
  size_t idx = (size_t)blockIdx.x * blockDim.x + threadIdx.x;
  if (idx >= (size_t)B_ * L_ * N_ * D_) return;
  int d = (int)(idx & 15);
  size_t t = idx >> 4;
  int n = (int)(t % N_);
  size_t bl = t / N_;
  int l = (int)(bl % L_);
  int b = (int)(bl / L_);
  int sidx = b * N_ + n;
  float xn = (x[((size_t)b * L_ + l) * N_ + n] - muv[sidx]) / stdv[sidx];
  float v = xn * sw[d] + sb[d];
  emb[((size_t)sidx * L_ + l) * D_ + d] = v;
  if (d == 0) g0[(size_t)sidx * L_ + l] = v;
}

// ---- K3: per-series trend (3 moving avgs) + top-4 Fourier seasonality ----
__global__ void k_gate_sig(const float* __restrict__ g0,
                           const float* __restrict__ glw, float* __restrict__ gl) {
  __shared__ float sg[L_];
  __shared__ float sXr[49], sXi[49], samp[49];
  __shared__ float thr_s;
  int s = blockIdx.x;
  int b = s / N_, n = s % N_;
  int t = threadIdx.x;
  sg[t] = g0[(size_t)s * L_ + t];
  __syncthreads();
  // trend: mean of moving averages k=4,8,12 with replicate padding
  float tr = 0.f;
  const int ks[3] = {4, 8, 12};
#pragma unroll
  for (int ki = 0; ki < 3; ++ki) {
    int k = ks[ki], f = (k - 1) / 2;
    float a = 0.f;
    for (int i = 0; i < k; ++i) {
      int c = t - f + i;
      c = c < 0 ? 0 : (c > L_ - 1 ? L_ - 1 : c);
      a += sg[c];
    }
    tr += a / k;
  }
  tr *= (1.f / 3.f);
  // naive rDFT (L=96 -> 49 bins)
  if (t < 49) {
    float re = 0.f, im = 0.f;
    for (int l = 0; l < L_; ++l) {
      float ph = -2.f * PI_F * (float)(t * l) / 96.f;
      re += sg[l] * __cosf(ph);
      im += sg[l] * __sinf(ph);
    }
    sXr[t] = re;
    sXi[t] = im;
    samp[t] = sqrtf(re * re + im * im);
  }
  __syncthreads();
  if (t == 0) {
    bool used[49];
    for (int k = 0; k < 49; ++k) used[k] = false;
    float v4 = 0.f;
    for (int it = 0; it < 4; ++it) {  // 4th largest (jax top_k tie semantics)
      float mx = -1e30f;
      int mi = 0;
      for (int k = 0; k < 49; ++k)
        if (!used[k] && samp[k] > mx) {
          mx = samp[k];
          mi = k;
        }
      used[mi] = true;
      v4 = mx;
    }
    thr_s = v4;
  }
  __syncthreads();
  // masked inverse rDFT
  float season = 0.f;
  for (int k = 0; k < 49; ++k) {
    if (samp[k] >= thr_s) {
      float wk = (k == 0 || k == 48) ? 1.f : 2.f;
      float ph = 2.f * PI_F * (float)(k * t) / 96.f;
      season += wk * (sXr[k] * __cosf(ph) - sXi[k] * __sinf(ph));
    }
  }
  season *= (1.f / 96.f);
  float sig = sg[t] + season + tr;
  atomicAdd(&gl[b * L_ + t], sig * glw[n]);
}

// ---- K4: logits, top-2 softmax gates, aux loss ----
__global__ void k_gate_fin(const float* __restrict__ gl, const float* __restrict__ glb,
                           const float* __restrict__ wgate, float* __restrict__ gates,
                           float* __restrict__ aux_out) {
  __shared__ float slog[B_ * E_];
  __shared__ float sg[B_ * E_];
  int tid = threadIdx.x;
  if (tid < B_ * E_) {
    int b = tid >> 2, e = tid & 3;
    float acc = 0.f;
    float bias = glb[0];
    for (int l = 0; l < L_; ++l) acc += (gl[b * L_ + l] + bias) * wgate[l * E_ + e];
    slog[tid] = acc;
  }
  __syncthreads();
  if (tid < B_) {
    int b = tid;
    float v[4];
    for (int e = 0; e < 4; ++e) v[e] = slog[b * 4 + e];
    int i1 = 0;
    for (int e = 1; e < 4; ++e)
      if (v[e] > v[i1]) i1 = e;
    int i2 = -1;
    for (int e = 0; e < 4; ++e) {
      if (e == i1) continue;
      if (i2 < 0 || v[e] > v[i2]) i2 = e;
    }
    float g1 = 1.f / (1.f + __expf(v[i2] - v[i1]));
    float g2 = 1.f - g1;
    for (int e = 0; e < 4; ++e) {
      float gg = (e == i1) ? g1 : ((e == i2) ? g2 : 0.f);
      sg[b * 4 + e] = gg;
      gates[b * 4 + e] = gg;
    }
  }
  __syncthreads();
  if (tid == 0) {
    float imp[4], ld[4];
    for (int e = 0; e < 4; ++e) {
      float si = 0.f, sl = 0.f;
      for (int b = 0; b < B_; ++b) {
        float gg = sg[b * 4 + e];
        si += gg;
        if (gg > 0.f) sl += 1.f;
      }
      imp[e] = si;
      ld[e] = sl;
    }
    float aux = 0.f;
    for (int which = 0; which < 2; ++which) {
      float* a = which ? ld : imp;
      float mean = (a[0] + a[1] + a[2] + a[3]) * 0.25f;
      float var = 0.f;
      for (int e = 0; e < 4; ++e) {
        float dv = a[e] - mean;
        var += dv * dv;
      }
      var *= (1.f / 3.f);  // ddof = 1
      aux += var / (mean * mean + 1e-10f);
    }
    aux_out[0] = aux * 1e-2f;
  }
}

// ---- K5: all experts for one (b,n) series; 6 waves, one 16x16 row tile each
__global__ void __launch_bounds__(192) k_experts(
    const float* __restrict__ emb, float* __restrict__ res,
    const float* __restrict__ gates, const float* __restrict__ Wq,
    const float* __restrict__ Wk, const float* __restrict__ Wv,
    const float* __restrict__ Wo, const float* __restrict__ Wq2,
    const float* __restrict__ Wk2, const float* __restrict__ Wv2,
    const float* __restrict__ Wo2, const float* __restrict__ W1,
    const float* __restrict__ b1, const float* __restrict__ W2,
    const float* __restrict__ b2) {
  __shared__ float sX[L_ * TS];    // emb tile [l][d]
  __shared__ float sH[L_ * TS];    // h after intra-attn; later h3
  __shared__ float sH2[L_ * TS];   // h2
  __shared__ float sP[48 * TS];    // pooled (padded to Tm*16 rows)
  __shared__ float sK2[48 * TS];
  __shared__ float sV2[48 * TS];
  __shared__ float sPO[48 * TS];   // inter-attn output
  __shared__ float sWS[6 * 4 * 16 * TS];  // 4 scratch tiles per wave

  const int bid = blockIdx.x;
  const int b = bid / N_;
  const int tid = threadIdx.x;
  const int lane = tid & 31;
  const int w = tid >> 5;
  float* WS = &sWS[w * (4 * 16 * TS)];

  for (int i = tid; i < LD; i += 192)
    sX[(i >> 4) * TS + (i & 15)] = emb[(size_t)bid * LD + i];
  __syncthreads();

  float racc[8];  // combined + emb residual, 8 elements per thread
#pragma unroll
  for (int j = 0; j < 8; ++j) {
    int idx = tid + 192 * j;
    racc[j] = sX[(idx >> 4) * TS + (idx & 15)];
  }

  const int patch[4] = {16, 8, 4, 2};
  for (int e = 0; e < E_; ++e) {
    const float g = gates[b * E_ + e];
    if (g != 0.f) {  // block-uniform: skip zero-gated experts
      const int p = patch[e];
      const int m = L_ / p;
      const int Tm = (m + 15) >> 4;
      // ---------- phase 1: intra-patch attention, wave w -> tile t = w
      {
        const int t = w;
        const float* Xt = &sX[t * 16 * TS];
        float* sQ = WS;
        float* sKt = WS + 272;
        float* sVt = WS + 544;
        float* sS = WS + 816;
        v8f q = {};
        q = mm16(Xt, TS, Wq + e * 256, 16, false, q, lane);
        st16(sQ, TS, q, lane, 1.f);
        v8f kk = {};
        kk = mm16(Xt, TS, Wk + e * 256, 16, false, kk, lane);
        st16(sKt, TS, kk, lane, 1.f);
        v8f vv = {};
        vv = mm16(Xt, TS, Wv + e * 256, 16, false, vv, lane);
        st16(sVt, TS, vv, lane, 1.f);
        v8f sc = {};
        sc = mm16(sQ, TS, sKt, TS, true, sc, lane);  // Q K^T (LDS transpose)
        st16(sS, TS, sc, lane, 0.25f);               // * 1/sqrt(16)
        if (lane < 16) {  // block-diagonal masked softmax per patch
          int r = lane;
          int j0 = (r / p) * p, j1 = j0 + p;
          float mx = -1e30f;
          for (int j = j0; j < j1; ++j) mx = fmaxf(mx, sS[r * TS + j]);
          float sm = 0.f;
          for (int j = j0; j < j1; ++j) {
            float ev = __expf(sS[r * TS + j] - mx);
            sS[r * TS + j] = ev;
            sm += ev;
          }
          float inv = 1.f / sm;
          for (int j = 0; j < 16; ++j)
            sS[r * TS + j] = (j >= j0 && j < j1) ? sS[r * TS + j] * inv : 0.f;
        }
        v8f av = {};
        av = mm16(sS, TS, sVt, TS, false, av, lane);
        st16(sQ, TS, av, lane, 1.f);  // reuse slot
        v8f o = {};
        o = mm16(sQ, TS, Wo + e * 256, 16, false, o, lane);
        {
          int cN = lane & 15, hi = lane >> 4;
#pragma unroll
          for (int r = 0; r < 8; ++r) o[r] += sX[(t * 16 + r + 8 * hi) * TS + cN];
        }
        st16(sS, TS, o, lane, 1.f);
        if (lane < 16) {  // LayerNorm rows -> sH
          int r = lane;
          float mu = 0.f;
          for (int j = 0; j < 16; ++j) mu += sS[r * TS + j];
          mu *= (1.f / 16.f);
          float var = 0.f;
          for (int j = 0; j < 16; ++j) {
            float dv = sS[r * TS + j] - mu;
            var += dv * dv;
          }
          var *= (1.f / 16.f);
          float inv = rsqrtf(var + 1e-5f);
          for (int j = 0; j < 16; ++j)
            sH[(t * 16 + r) * TS + j] = (sS[r * TS + j] - mu) * inv;
        }
      }
      __syncthreads();
      // pooled[mi][d] = mean over patch, zero-padded to Tm*16 rows
      for (int i = tid; i < Tm * 16 * 16; i += 192) {
        int mi = i >> 4, d = i & 15;
        float sacc = 0.f;
        if (mi < m) {
          for (int q = 0; q < p; ++q) sacc += sH[(mi * p + q) * TS + d];
          sacc /= (float)p;
        }
        sP[mi * TS + d] = sacc;
      }
      __syncthreads();
      // ---------- phase 2: inter-patch attention over m rows (Tm tiles)
      if (w < Tm) {
        const int t = w;
        v8f kk = {};
        kk = mm16(&sP[t * 16 * TS], TS, Wk2 + e * 256, 16, false, kk, lane);
        st16(&sK2[t * 16 * TS], TS, kk, lane, 1.f);
        v8f vv = {};
        vv = mm16(&sP[t * 16 * TS], TS, Wv2 + e * 256, 16, false, vv, lane);
        st16(&sV2[t * 16 * TS], TS, vv, lane, 1.f);
        v8f q2 = {};
        q2 = mm16(&sP[t * 16 * TS], TS, Wq2 + e * 256, 16, false, q2, lane);
        st16(WS, TS, q2, lane, 1.f);
      }
      __syncthreads();
      if (w < Tm) {
        const int t = w;
        for (int ct = 0; ct < Tm; ++ct) {
          v8f sc = {};
          sc = mm16(WS, TS, &sK2[ct * 16 * TS], TS, true, sc, lane);
          st16(WS + (1 + ct) * 272, TS, sc, lane, 0.25f);
        }
        if (lane < 16) {  // softmax over m valid cols
          int r = lane;
          float mx = -1e30f;
          for (int c = 0; c < m; ++c)
            mx = fmaxf(mx, WS[(1 + (c >> 4)) * 272 + r * TS + (c & 15)]);
          float sm = 0.f;
          for (int c = 0; c < m; ++c) {
            float* pe = &WS[(1 + (c >> 4)) * 272 + r * TS + (c & 15)];
            float ev = __expf(*pe - mx);
            *pe = ev;
            sm += ev;
          }
          float inv = 1.f / sm;
          for (int c = 0; c < Tm * 16; ++c) {
            float* pe = &WS[(1 + (c >> 4)) * 272 + r * TS + (c & 15)];
            *pe = (c < m) ? (*pe) * inv : 0.f;
          }
        }
        v8f av = {};
        for (int ct = 0; ct < Tm; ++ct)
          av = mm16(WS + (1 + ct) * 272, TS, &sV2[ct * 16 * TS], TS, false, av, lane);
        st16(WS, TS, av, lane, 1.f);
        v8f o2 = {};
        o2 = mm16(WS, TS, Wo2 + e * 256, 16, false, o2, lane);
        st16(&sPO[t * 16 * TS], TS, o2, lane, 1.f);
      }
      __syncthreads();
      // h2 = LN(h + inter broadcast over patch)
      for (int l = tid; l < L_; l += 192) {
        int mi = l / p;
        float tmp[16];
        float mu = 0.f;
#pragma unroll
        for (int d = 0; d < 16; ++d) {
          tmp[d] = sH[l * TS + d] + sPO[mi * TS + d];
          mu += tmp[d];
        }
        mu *= (1.f / 16.f);
        float var = 0.f;
#pragma unroll
        for (int d = 0; d < 16; ++d) {
          float dv = tmp[d] - mu;
          var += dv * dv;
        }
        var *= (1.f / 16.f);
        float inv = rsqrtf(var + 1e-5f);
#pragma unroll
        for (int d = 0; d < 16; ++d) sH2[l * TS + d] = (tmp[d] - mu) * inv;
      }
      __syncthreads();
      // ---------- phase 3: FFN (16->64 gelu ->16) + residual + LN -> sH
      {
        const int t = w;
        const int cN = lane & 15, hi = lane >> 4;
        for (int c = 0; c < 4; ++c) {
          v8f ga = {};
          ga = mm16(&sH2[t * 16 * TS], TS, W1 + e * (D_ * F_) + c * 16, F_, false,
                    ga, lane);
          float bb = b1[e * F_ + c * 16 + cN];
#pragma unroll
          for (int r = 0; r < 8; ++r) ga[r] = gelu_f(ga[r] + bb);
          st16(WS + c * 272, TS, ga, lane, 1.f);
        }
        v8f f2 = {};
        for (int c = 0; c < 4; ++c)
          f2 = mm16(WS + c * 272, TS, W2 + e * (F_ * D_) + c * 16 * D_, D_, false,
                    f2, lane);
        float bb2 = b2[e * D_ + cN];
#pragma unroll
        for (int r = 0; r < 8; ++r)
          f2[r] += bb2 + sH2[(t * 16 + r + 8 * hi) * TS + cN];
        st16(WS, TS, f2, lane, 1.f);
        if (lane < 16) {  // final LN -> h3 into sH
          int r = lane;
          float mu = 0.f;
          for (int j = 0; j < 16; ++j) mu += WS[r * TS + j];
          mu *= (1.f / 16.f);
          float var = 0.f;
          for (int j = 0; j < 16; ++j) {
            float dv = WS[r * TS + j] - mu;
            var += dv * dv;
          }
          var *= (1.f / 16.f);
          float inv = rsqrtf(var + 1e-5f);
          for (int j = 0; j < 16; ++j)
            sH[(t * 16 + r) * TS + j] = (WS[r * TS + j] - mu) * inv;
        }
      }
      __syncthreads();
#pragma unroll
      for (int j = 0; j < 8; ++j) {
        int idx = tid + 192 * j;
        racc[j] += g * sH[(idx >> 4) * TS + (idx & 15)];
      }
      __syncthreads();
    }
  }
#pragma unroll
  for (int j = 0; j < 8; ++j) {
    int idx = tid + 192 * j;
    res[(size_t)bid * LD + idx] = racc[j];  // res layout == [B,N,L*D]
  }
}

// ---- K6: final projection [16 rows x 96 cols x K=1536] per block + denorm --
__global__ void __launch_bounds__(256) k_proj(
    const float* __restrict__ res, const float* __restrict__ pw,
    const float* __restrict__ pb, const float* __restrict__ muv,
    const float* __restrict__ stdv, float* __restrict__ out) {
  __shared__ float sA[16 * TS];
  __shared__ float sRow[16];
  const int tid = threadIdx.x;
  const int lane = tid & 31;
  const int w = tid >> 5;  // waves 0..5 own col tiles; 6,7 just help load
  const int R0 = blockIdx.x * 16;
  const int ar = tid >> 4, ac = tid & 15;
  if (tid < 16) sRow[tid] = 0.f;
  float rsum = 0.f;
  v8f acc = {};
  for (int kc = 0; kc < LD / 16; ++kc) {
    __syncthreads();
    float v = res[(size_t)(R0 + ar) * LD + kc * 16 + ac];
    sA[ar * TS + ac] = v;
    rsum += v;
    __syncthreads();
    if (w < 6)
      acc = mm16(sA, TS, pw + (size_t)kc * 16 * P_ + w * 16, P_, false, acc, lane);
  }
  atomicAdd(&sRow[ar], rsum);
  __syncthreads();
  if (w < 6) {
    const int cN = lane & 15, hi = lane >> 4;
    const int jcol = w * 16 + cN;
#pragma unroll
    for (int r = 0; r < 8; ++r) {
      int row = r + 8 * hi;
      int R = R0 + row;
      int bb = R / N_, nn = R % N_;
      float val = acc[r] + pb[jcol] + sRow[row] * (1.f / (float)LD);
      val = val * stdv[R] + muv[R];  // RevIN denorm
      out[((size_t)bb * P_ + jcol) * N_ + nn] = val;
    }
  }
}

extern "C" void kernel_launch(void* const* d_in, const int* in_sizes, int n_in,
                              void* d_out, int out_size, void* d_ws, size_t ws_size,
                              hipStream_t stream) {
  const float* x = (const float*)d_in[0];
  const float* sw = (const float*)d_in[1];
  const float* sb = (const float*)d_in[2];
  const float* glw = (const float*)d_in[3];
  const float* glb = (const float*)d_in[4];
  const float* wgate = (const float*)d_in[5];
  const float* Wq = (const float*)d_in[6];
  const float* Wk = (const float*)d_in[7];
  const float* Wv = (const float*)d_in[8];
  const float* Wo = (const float*)d_in[9];
  const float* Wq2 = (const float*)d_in[10];
  const float* Wk2 = (const float*)d_in[11];
  const float* Wv2 = (const float*)d_in[12];
  const float* Wo2 = (const float*)d_in[13];
  const float* W1 = (const float*)d_in[14];
  const float* b1 = (const float*)d_in[15];
  const float* W2 = (const float*)d_in[16];
  const float* b2 = (const float*)d_in[17];
  const float* pw = (const float*)d_in[18];
  const float* pb = (const float*)d_in[19];
  float* out = (float*)d_out;

  float* ws = (float*)d_ws;
  float* emb = ws;                          // [B,N,L,D]
  float* res = emb + (size_t)BN * LD;       // [B,N,L*D]
  float* g0 = res + (size_t)BN * LD;        // [B,N,L]
  float* muv = g0 + (size_t)BN * L_;
  float* stdv = muv + BN;
  float* gl = stdv + BN;                    // [B,L]
  float* gates = gl + B_ * L_;              // [B,E]

  hipMemsetAsync(gl, 0, B_ * L_ * sizeof(float), stream);

  k_stats<<<(BN + 255) / 256, 256, 0, stream>>>(x, muv, stdv);
  size_t tot = (size_t)B_ * L_ * N_ * D_;
  k_embed<<<(int)((tot + 255) / 256), 256, 0, stream>>>(x, muv, stdv, sw, sb, emb, g0);
  k_gate_sig<<<BN, L_, 0, stream>>>(g0, glw, gl);
  k_gate_fin<<<1, 64, 0, stream>>>(gl, glb, wgate, gates,
                                   out + (size_t)B_ * P_ * N_);
  k_experts<<<BN, 192, 0, stream>>>(emb, res, gates, Wq, Wk, Wv, Wo, Wq2, Wk2,
                                    Wv2, Wo2, W1, b1, W2, b2);
  k_proj<<<BN / 16, 256, 0, stream>>>(res, pw, pb, muv, stdv, out);
}